// SVGDModel_48387101557205
// MI455X (gfx1250) — compile-verified
//
#include <hip/hip_runtime.h>

// SVGD gradient, fused:  phi[i] = ( rowsum_i*(mu + x_i) - 2*(G@X)_i ) / N
// G[i,j] = exp(-||x_i - x_j||^2 / 2) computed via S = X X^T on the f32 WMMA pipe.

typedef __attribute__((ext_vector_type(2))) float v2f;
typedef __attribute__((ext_vector_type(8))) float v8f;

#define N_PART 4096
#define DIMS   16
#define LOG2E  1.44269504088896340736f

// ---------------------------------------------------------------------------
// Kernel 1: per-particle squared norms (4096 floats -> d_ws)
// ---------------------------------------------------------------------------
__global__ __launch_bounds__(256) void svgd_norms_kernel(const float* __restrict__ X,
                                                         float* __restrict__ norms) {
    int i = blockIdx.x * blockDim.x + threadIdx.x;   // 0..4095
    const float4* r = (const float4*)(X + (size_t)i * DIMS);
    float s = 0.f;
#pragma unroll
    for (int q = 0; q < 4; ++q) {
        float4 v = r[q];
        s += v.x * v.x + v.y * v.y + v.z * v.z + v.w * v.w;
    }
    norms[i] = s;
}

// ---------------------------------------------------------------------------
// Kernel 2: fused SVGD phi. One wave (32 threads) owns a 16-row i-tile and
// sweeps all 4096 j in 16-wide tiles. Per j-tile:
//   4x v_wmma_f32_16x16x4_f32  (S = Xi . Xj^T)
//   8x v_exp_f32               (G tile)
//   LDS C-layout -> A-layout transpose of G
//   4x v_wmma_f32_16x16x4_f32  (GX += G . Xj)
// ---------------------------------------------------------------------------
__global__ __launch_bounds__(32) void svgd_phi_kernel(const float* __restrict__ X,
                                                      const float* __restrict__ mu,
                                                      const float* __restrict__ norms,
                                                      float* __restrict__ out) {
    // 16x16 G tile staged with stride 18 floats:
    //  - ds_store_b32 writes (i_local*18 + j_local): conflict-free
    //  - ds_load_b64  reads (row*18 + 4kk + 2hi): 8B-aligned, conflict-free
    __shared__ float gtile[16 * 18];

    const int lane = threadIdx.x;    // 0..31 (wave32)
    const int row  = lane & 15;
    const int hi   = lane >> 4;      // 0 or 1
    const int i0   = blockIdx.x * 16;

    // A-operand chunks of the Xi tile (16x16 split into 4 K=4 chunks).
    // f32 16x4 A layout: lanes 0-15 hold K={0,1}, lanes 16-31 hold K={2,3}.
    v2f aS[4];
#pragma unroll
    for (int kk = 0; kk < 4; ++kk)
        aS[kk] = *(const v2f*)(X + (size_t)(i0 + row) * DIMS + 4 * kk + 2 * hi);

    // Row norms for the 8 i-rows this lane's C-slots map to (M = v + 8*hi).
    float niv[8];
#pragma unroll
    for (int v = 0; v < 8; ++v)
        niv[v] = norms[i0 + v + 8 * hi];

    v8f gx = {0.f, 0.f, 0.f, 0.f, 0.f, 0.f, 0.f, 0.f};   // (G @ X) accumulator, C-layout
    float rs[8] = {0.f, 0.f, 0.f, 0.f, 0.f, 0.f, 0.f, 0.f}; // partial row sums of G

    for (int j0 = 0; j0 < N_PART; j0 += 16) {
        // B-operand of S = Xi . Xj^T : B[k][n] = X[j0+n][k]; mirrored per-lane
        // pattern means the load is identical in form to the A load.
        v2f bS[4];
#pragma unroll
        for (int kk = 0; kk < 4; ++kk)
            bS[kk] = *(const v2f*)(X + (size_t)(j0 + row) * DIMS + 4 * kk + 2 * hi);

        v8f s = {0.f, 0.f, 0.f, 0.f, 0.f, 0.f, 0.f, 0.f};
#pragma unroll
        for (int kk = 0; kk < 4; ++kk)
            s = __builtin_amdgcn_wmma_f32_16x16x4_f32(
                    false, aS[kk], false, bS[kk], (short)0, s, false, false);

        // G = exp(S - 0.5*(ni + nj)) = exp2((S - 0.5*(ni+nj)) * log2(e))
        const float nj = norms[j0 + row];   // N = row in the C layout
        v8f g;
#pragma unroll
        for (int v = 0; v < 8; ++v) {
            g[v] = __builtin_amdgcn_exp2f((s[v] - 0.5f * (niv[v] + nj)) * LOG2E);
            rs[v] += g[v];
        }

        // C-layout -> LDS (i_local, j_local) -> A-layout for the second GEMM.
#pragma unroll
        for (int v = 0; v < 8; ++v)
            gtile[(v + 8 * hi) * 18 + row] = g[v];
        // Same-wave LDS store->load: in-order, dependency visible to compiler.
        v2f aG[4];
#pragma unroll
        for (int kk = 0; kk < 4; ++kk)
            aG[kk] = *(const v2f*)&gtile[row * 18 + 4 * kk + 2 * hi];

        // B-operand of GX += G . Xj : B[k][n] = X[j0+k][n], n = d = row.
        v2f bX[4];
#pragma unroll
        for (int kk = 0; kk < 4; ++kk) {
            const int jr = j0 + 4 * kk + 2 * hi;
            v2f b = { X[(size_t)jr * DIMS + row], X[(size_t)(jr + 1) * DIMS + row] };
            bX[kk] = b;
        }
#pragma unroll
        for (int kk = 0; kk < 4; ++kk)
            gx = __builtin_amdgcn_wmma_f32_16x16x4_f32(
                    false, aG[kk], false, bX[kk], (short)0, gx, false, false);
    }

    // Finish rowsums: reduce over the 16 lanes of each half (xor stays in-group).
#pragma unroll
    for (int v = 0; v < 8; ++v) {
        rs[v] += __shfl_xor(rs[v], 1, 32);
        rs[v] += __shfl_xor(rs[v], 2, 32);
        rs[v] += __shfl_xor(rs[v], 4, 32);
        rs[v] += __shfl_xor(rs[v], 8, 32);
    }

    // phi[i,d] = (rowsum_i*(mu[d] + x[i,d]) - 2*GX[i,d]) / N ; d = row, i = i0+v+8*hi.
    const float mud = mu[row];
#pragma unroll
    for (int v = 0; v < 8; ++v) {
        const int i = i0 + v + 8 * hi;
        const float xid = X[(size_t)i * DIMS + row];
        out[(size_t)i * DIMS + row] =
            (rs[v] * (mud + xid) - 2.0f * gx[v]) * (1.0f / (float)N_PART);
    }
}

// ---------------------------------------------------------------------------
extern "C" void kernel_launch(void* const* d_in, const int* in_sizes, int n_in,
                              void* d_out, int out_size, void* d_ws, size_t ws_size,
                              hipStream_t stream) {
    (void)in_sizes; (void)n_in; (void)out_size; (void)ws_size;
    const float* particles = (const float*)d_in[0];   // [4096,16] f32
    const float* mu        = (const float*)d_in[1];   // [16] f32
    float*       out       = (float*)d_out;           // [4096,16] f32
    float*       norms     = (float*)d_ws;            // 4096 f32 = 16 KB scratch

    svgd_norms_kernel<<<N_PART / 256, 256, 0, stream>>>(particles, norms);
    svgd_phi_kernel<<<N_PART / 16, 32, 0, stream>>>(particles, mu, norms, out);
}